// MoEFeedForward_69398081569309
// MI455X (gfx1250) — compile-verified
//
#include <hip/hip_runtime.h>
#include <hip/hip_bf16.h>

typedef __attribute__((ext_vector_type(2))) float v2f;
typedef __attribute__((ext_vector_type(8))) float v8f;

#define Bn 4
#define Sn 4096
#define Hn 2048
#define NVn 5000
#define NGn (NVn + 1)               // 5001 gate slots (last = identity)
#define Ln 33
#define LAYERn 16
#define NTILES ((NGn + 15) / 16)    // 313 N-tiles of 16 gates
#define NPAIRS ((NTILES + 1) / 2)   // 157 pairs of N-tiles (32 gates each)
#define MW 32                       // tokens per workgroup (fp32 A fits 320KB LDS)
#define LDS_STRIDE (Hn + 4)         // 2052 floats: conflict-free ds b64 A-fetch

#define WMMA_F32(A, Bv, C) \
    __builtin_amdgcn_wmma_f32_16x16x4_f32(false, (A), false, (Bv), (short)0, (C), false, false)

// ---------------------------------------------------------------------------
// Kernel 1: fused GEMM + per-token argmax.
// grid.x = (B*S)/32, block = 256 (8 wave32). Each wave owns a 32M x 32N
// register tile: 4 independent WMMA accumulator chains, 2 LDS-A + 2 global-B
// b64 loads per 4 WMMAs. Scores are never materialized; running (max,idx)
// is kept in registers and reduced via ds_bpermute shuffles + LDS.
// ---------------------------------------------------------------------------
__global__ void gemm_argmax_kernel(const float* __restrict__ x,
                                   const float* __restrict__ gw,
                                   const float* __restrict__ gb,
                                   int* __restrict__ out_idx) {
    extern __shared__ float lds[];                 // [32][LDS_STRIDE] x-block
    float* redV = lds + MW * LDS_STRIDE;           // [8 waves][32 tokens]
    int*   redI = (int*)(redV + 8 * MW);           // [8 waves][32 tokens]

    const int tid  = threadIdx.x;
    const int wave = tid >> 5;
    const int lane = tid & 31;
    const int half = lane >> 4;     // selects K pair {0,1} vs {2,3}
    const int l15  = lane & 15;     // M row (A) / N col (B)

    const long long rowBase = (long long)blockIdx.x * MW;

    // Stage 32x2048 fp32 x-block into LDS (float4 cooperative copy).
    for (int i = tid; i < MW * (Hn / 4); i += 256) {
        const int r  = i / (Hn / 4);
        const int c4 = i % (Hn / 4);
        const float4 v = ((const float4*)(x + (rowBase + r) * Hn))[c4];
        float* dst = lds + r * LDS_STRIDE + c4 * 4;
        dst[0] = v.x; dst[1] = v.y; dst[2] = v.z; dst[3] = v.w;
    }
    __syncthreads();

    float best0V[8], best1V[8];     // running max for M-tile 0 / 1
    int   best0I[8], best1I[8];
#pragma unroll
    for (int r = 0; r < 8; ++r) {
        best0V[r] = -__builtin_inff(); best0I[r] = 0;
        best1V[r] = -__builtin_inff(); best1I[r] = 0;
    }

    const float* aRow0 = lds + l15 * LDS_STRIDE + 2 * half;  // M rows 0..15
    const float* aRow1 = aRow0 + 16 * LDS_STRIDE;            // M rows 16..31

    for (int p = wave; p < NPAIRS; p += 8) {
        const int n0 = p * 32 + l15;            // this lane's gate, N-tile 0
        const int n1 = n0 + 16;                 // this lane's gate, N-tile 1
        const float* b0 = gw + (long long)(n0 < NGn ? n0 : NGn - 1) * Hn + 2 * half;
        const float* b1 = gw + (long long)(n1 < NGn ? n1 : NGn - 1) * Hn + 2 * half;

        v8f a00 = {0,0,0,0,0,0,0,0}, a01 = {0,0,0,0,0,0,0,0};
        v8f a10 = {0,0,0,0,0,0,0,0}, a11 = {0,0,0,0,0,0,0,0};
#pragma unroll 4
        for (int k = 0; k < Hn; k += 4) {
            const v2f A0 = *(const v2f*)(aRow0 + k);   // ds_load_b64
            const v2f A1 = *(const v2f*)(aRow1 + k);   // ds_load_b64
            const v2f B0 = *(const v2f*)(b0 + k);      // global_load_b64
            const v2f B1 = *(const v2f*)(b1 + k);      // global_load_b64
            a00 = WMMA_F32(A0, B0, a00);               // 4 independent chains
            a01 = WMMA_F32(A0, B1, a01);
            a10 = WMMA_F32(A1, B0, a10);
            a11 = WMMA_F32(A1, B1, a11);
        }

        const float bias0 = (n0 < NGn) ? gb[n0] : 0.f;
        const float bias1 = (n1 < NGn) ? gb[n1] : 0.f;
#pragma unroll
        for (int r = 0; r < 8; ++r) {
            // n0 before n1, strict '>' => lowest index wins ties (jnp.argmax)
            float s = (n0 < NGn) ? (a00[r] + bias0) : -__builtin_inff();
            if (s > best0V[r]) { best0V[r] = s; best0I[r] = n0; }
            s = (n1 < NGn) ? (a01[r] + bias1) : -__builtin_inff();
            if (s > best0V[r]) { best0V[r] = s; best0I[r] = n1; }
            s = (n0 < NGn) ? (a10[r] + bias0) : -__builtin_inff();
            if (s > best1V[r]) { best1V[r] = s; best1I[r] = n0; }
            s = (n1 < NGn) ? (a11[r] + bias1) : -__builtin_inff();
            if (s > best1V[r]) { best1V[r] = s; best1I[r] = n1; }
        }
    }

    // Reduce across the 16 lanes of each half (C layout: VGPR r, half -> M).
#pragma unroll
    for (int r = 0; r < 8; ++r) {
        float v0 = best0V[r], v1 = best1V[r];
        int   i0 = best0I[r], i1 = best1I[r];
#pragma unroll
        for (int m = 8; m >= 1; m >>= 1) {
            const float ov0 = __shfl_xor(v0, m, 32);
            const int   oi0 = __shfl_xor(i0, m, 32);
            if (ov0 > v0 || (ov0 == v0 && oi0 < i0)) { v0 = ov0; i0 = oi0; }
            const float ov1 = __shfl_xor(v1, m, 32);
            const int   oi1 = __shfl_xor(i1, m, 32);
            if (ov1 > v1 || (ov1 == v1 && oi1 < i1)) { v1 = ov1; i1 = oi1; }
        }
        best0V[r] = v0; best0I[r] = i0;
        best1V[r] = v1; best1I[r] = i1;
    }

    // Lanes 0 and 16 hold results for tokens (half*8+r) and 16+(half*8+r).
    if (l15 == 0) {
#pragma unroll
        for (int r = 0; r < 8; ++r) {
            redV[wave * MW + half * 8 + r]      = best0V[r];
            redI[wave * MW + half * 8 + r]      = best0I[r];
            redV[wave * MW + 16 + half * 8 + r] = best1V[r];
            redI[wave * MW + 16 + half * 8 + r] = best1I[r];
        }
    }
    __syncthreads();

    // Combine the 8 waves' N-subsets; threads 0..31 own one token each.
    if (tid < MW) {
        float v = redV[tid];
        int   i = redI[tid];
        for (int w = 1; w < 8; ++w) {
            const float ov = redV[w * MW + tid];
            const int   oi = redI[w * MW + tid];
            if (ov > v || (ov == v && oi < i)) { v = ov; i = oi; }
        }
        out_idx[rowBase + tid] = i;
    }
}

// ---------------------------------------------------------------------------
// Kernel 2: pure-bandwidth gather/select. One block per token.
// ---------------------------------------------------------------------------
__global__ void gather_out_kernel(const float* __restrict__ x,
                                  const float* __restrict__ pool,
                                  const int* __restrict__ idx,
                                  float* __restrict__ out) {
    const long long token = blockIdx.x;
    const int tid = threadIdx.x;
    const int id = idx[token];

    const float* src;
    if (id == NVn) {
        src = x + token * Hn;                               // identity slot
    } else {
        const int s = (id < NVn - 1) ? id : (NVn - 1);      // safe_idx
        src = pool + ((long long)s * Ln + LAYERn) * Hn;
    }

    float4* o4 = (float4*)(out + token * Hn);
    const float4* s4 = (const float4*)src;
    for (int i = tid; i < Hn / 4; i += 256) o4[i] = s4[i];
}

extern "C" void kernel_launch(void* const* d_in, const int* in_sizes, int n_in,
                              void* d_out, int out_size, void* d_ws, size_t ws_size,
                              hipStream_t stream) {
    (void)in_sizes; (void)n_in; (void)out_size; (void)ws_size;
    const float* x    = (const float*)d_in[0];
    const float* gw   = (const float*)d_in[1];
    const float* gb   = (const float*)d_in[2];
    const float* pool = (const float*)d_in[3];
    int* idx = (int*)d_ws;                         // 16384 ints of scratch

    const size_t shmem = (MW * LDS_STRIDE + 8 * MW) * sizeof(float)
                       + 8 * MW * sizeof(int);     // ~265 KB (CDNA5: 320 KB/WGP)

    gemm_argmax_kernel<<<(Bn * Sn) / MW, 256, shmem, stream>>>(x, gw, gb, idx);
    gather_out_kernel<<<Bn * Sn, 256, 0, stream>>>(x, pool, idx, (float*)d_out);
}